// RelGraphConvHeteroEmbed_76501957476383
// MI455X (gfx1250) — compile-verified
//
#include <hip/hip_runtime.h>
#include <hip/hip_bf16.h>

#define NA_N 50000
#define NB_N 50000
#define E_N  400000
#define DIM  128

typedef __attribute__((ext_vector_type(2))) float v2f;
typedef __attribute__((ext_vector_type(8))) float v8f;

// ---------------------------------------------------------------------------
// Zero-fill (grid-stride). d_out and the count scratch must be zeroed every
// call (harness poisons them and never re-poisons between replays).
// ---------------------------------------------------------------------------
__global__ void zero_f32(float* __restrict__ p, int n) {
    int i = blockIdx.x * blockDim.x + threadIdx.x;
    int stride = gridDim.x * blockDim.x;
    for (; i < n; i += stride) p[i] = 0.0f;
}

// ---------------------------------------------------------------------------
// Pass 1: per-etype in-degree counts (float, since we divide later).
// ---------------------------------------------------------------------------
__global__ void edge_count(const int* __restrict__ dst,
                           float* __restrict__ cnt, int E) {
    int i = blockIdx.x * blockDim.x + threadIdx.x;
    if (i < E) atomicAdd(&cnt[dst[i]], 1.0f);
}

// ---------------------------------------------------------------------------
// Pass 2: scatter  out[dst] += embed[src] / max(cnt[dst],1).
// Block = 256 threads = 8 waves. The block stages its 256 (src,dst) pairs
// into LDS with the CDNA5 async global->LDS path (ASYNCcnt-tracked), then
// each wave owns 32 edges. D=128 f32 maps exactly onto 32 lanes x float4:
// one wave touches one full row per edge with a single b128 load per lane.
// ---------------------------------------------------------------------------
__global__ __launch_bounds__(256) void edge_scatter(
    const float* __restrict__ embed,   // [Nsrc, DIM]
    const int*   __restrict__ src,
    const int*   __restrict__ dst,
    const float* __restrict__ cnt,     // per-etype dst counts
    float*       __restrict__ out,     // [Ndst, DIM] accumulator
    int E)
{
    __shared__ int s_src[256];
    __shared__ int s_dst[256];

    const int tid  = threadIdx.x;
    const int lane = tid & 31;
    const int wave = tid >> 5;
    const int base = blockIdx.x * 256;
    const int gid  = base + tid;

    // Async index staging: global -> LDS without VGPR round-trip.
    {
        unsigned lds_s = (unsigned)(size_t)&s_src[tid];
        unsigned lds_d = (unsigned)(size_t)&s_dst[tid];
        unsigned goff  = (unsigned)gid * 4u;
        if (gid < E) {
            asm volatile("global_load_async_to_lds_b32 %0, %1, %2"
                         :: "v"(lds_s), "v"(goff), "s"(src) : "memory");
            asm volatile("global_load_async_to_lds_b32 %0, %1, %2"
                         :: "v"(lds_d), "v"(goff), "s"(dst) : "memory");
        }
    }
#if __has_builtin(__builtin_amdgcn_s_wait_asynccnt)
    __builtin_amdgcn_s_wait_asynccnt(0);
#else
    asm volatile("s_wait_asynccnt 0x0" ::: "memory");
#endif
    __syncthreads();

    const int ebase = wave * 32;
#pragma unroll 1
    for (int e = 0; e < 32; ++e) {
        const int eg = base + ebase + e;        // uniform across the wave
        if (eg >= E) break;
        const int s = s_src[ebase + e];         // LDS broadcast read
        const int d = s_dst[ebase + e];

        // Prefetch next edge's source row (global_prefetch_b8) to hide
        // the random-gather latency of the serial per-wave edge loop.
        if (e + 1 < 32 && eg + 1 < E) {
            const int sn = s_src[ebase + e + 1];
            __builtin_prefetch(&embed[(size_t)sn * DIM + lane * 4], 0, 0);
        }

        const float c = cnt[d];                 // same addr all lanes -> 1 req
        const float r = 1.0f / fmaxf(c, 1.0f);  // mean as pre-scaled sum

        const float4 v = *(const float4*)(embed + (size_t)s * DIM + lane * 4);
        float* o = out + (size_t)d * DIM + lane * 4;
        atomicAdd(o + 0, v.x * r);
        atomicAdd(o + 1, v.y * r);
        atomicAdd(o + 2, v.z * r);
        atomicAdd(o + 3, v.w * r);
    }
}

// ---------------------------------------------------------------------------
// Epilogue: acc = relu(acc + self + bias_broadcast), one wave per 16x16 tile
// in the native WMMA C/D layout (lane 0-15 -> N, VGPR v -> M=v / M=v+8).
// The rank-1 bias broadcast ones(16) (x) bias(16) runs on the matrix pipe:
//   D = A(16x4) x B(4x16) + C,  A[:,0]=1, B[0,:]=bias  =>  D = C + bias[n].
// EXEC is all-1s (exact tiling: 50000 rows = 3125*16, 128 cols = 8 waves),
// satisfying the WMMA EXEC restriction. RNE add == VALU add numerics.
// ---------------------------------------------------------------------------
__global__ __launch_bounds__(256) void epilogue_wmma(
    float* __restrict__ acc,
    const float* __restrict__ self_embed,
    const float* __restrict__ bias)
{
    const int lane = threadIdx.x & 31;
    const int wave = threadIdx.x >> 5;      // column tile 0..7
    const int row0 = blockIdx.x * 16;
    const int col0 = wave * 16;
    const int n    = lane & 15;
    const int hi   = lane >> 4;             // 0: M=v, 1: M=v+8

    // C = acc + self for this tile, in C/D register layout.
    v8f c;
#pragma unroll
    for (int v = 0; v < 8; ++v) {
        const size_t idx = (size_t)(row0 + v + 8 * hi) * DIM + col0 + n;
        c[v] = acc[idx] + self_embed[idx];
    }

    const float bv = bias[col0 + n];

#if __has_builtin(__builtin_amdgcn_wmma_f32_16x16x4_f32)
    // A (16x4 f32): lanes 0-15 hold K=0 (VGPR0) / K=1 (VGPR1); lanes 16-31
    // hold K=2/K=3.  Ones in column K=0 only.
    v2f a, b;
    a.x = (hi == 0) ? 1.0f : 0.0f;
    a.y = 0.0f;
    // B (4x16 f32): row K=0 lives in VGPR0 lanes 0-15. bias there, else 0.
    b.x = (hi == 0) ? bv : 0.0f;
    b.y = 0.0f;
    c = __builtin_amdgcn_wmma_f32_16x16x4_f32(false, a, false, b,
                                              (short)0, c, false, false);
#else
#pragma unroll
    for (int v = 0; v < 8; ++v) c[v] += bv;
#endif

#pragma unroll
    for (int v = 0; v < 8; ++v) {
        const size_t idx = (size_t)(row0 + v + 8 * hi) * DIM + col0 + n;
        acc[idx] = fmaxf(c[v], 0.0f);
    }
}

// ---------------------------------------------------------------------------
extern "C" void kernel_launch(void* const* d_in, const int* in_sizes, int n_in,
                              void* d_out, int out_size, void* d_ws, size_t ws_size,
                              hipStream_t stream) {
    const float* embed_r0 = (const float*)d_in[0];   // [NA, D]
    const float* embed_r1 = (const float*)d_in[1];   // [NB, D]
    const float* embed_r2 = (const float*)d_in[2];   // [NA, D]
    const float* self_A   = (const float*)d_in[3];   // [NA, D]
    const float* self_B   = (const float*)d_in[4];   // [NB, D]
    const float* h_bias   = (const float*)d_in[5];   // [D]
    const int* src0 = (const int*)d_in[6];
    const int* dst0 = (const int*)d_in[7];
    const int* src1 = (const int*)d_in[8];
    const int* dst1 = (const int*)d_in[9];
    const int* src2 = (const int*)d_in[10];
    const int* dst2 = (const int*)d_in[11];

    float* hA = (float*)d_out;                       // [NA, D]
    float* hB = (float*)d_out + (size_t)NA_N * DIM;  // [NB, D]

    float* cnt0 = (float*)d_ws;                      // [NB]  (etype r0: A->B)
    float* cnt1 = cnt0 + NB_N;                       // [NA]  (etype r1: B->A)
    float* cnt2 = cnt1 + NA_N;                       // [NA]  (etype r2: A->A)
    const int ncnt = NB_N + NA_N + NA_N;

    const int TB = 256;
    const int nOut = (NA_N + NB_N) * DIM;

    // Zero accumulators + counts (on-stream, capture-safe).
    zero_f32<<<4096, TB, 0, stream>>>((float*)d_out, nOut);
    zero_f32<<<(ncnt + TB - 1) / TB, TB, 0, stream>>>(cnt0, ncnt);

    const int EB = (E_N + TB - 1) / TB;

    // Pass 1: per-etype in-degree counts.
    edge_count<<<EB, TB, 0, stream>>>(dst0, cnt0, E_N);
    edge_count<<<EB, TB, 0, stream>>>(dst1, cnt1, E_N);
    edge_count<<<EB, TB, 0, stream>>>(dst2, cnt2, E_N);

    // Pass 2: pre-scaled scatter-sum (== per-etype mean, summed per ntype).
    edge_scatter<<<EB, TB, 0, stream>>>(embed_r0, src0, dst0, cnt0, hB, E_N);
    edge_scatter<<<EB, TB, 0, stream>>>(embed_r1, src1, dst1, cnt1, hA, E_N);
    edge_scatter<<<EB, TB, 0, stream>>>(embed_r2, src2, dst2, cnt2, hA, E_N);

    // Epilogue: relu(acc + self + bias), in place in d_out.
    epilogue_wmma<<<NA_N / 16, TB, 0, stream>>>(hA, self_A, h_bias);
    epilogue_wmma<<<NB_N / 16, TB, 0, stream>>>(hB, self_B, h_bias);
}